// NewNorm_11811160064499
// MI455X (gfx1250) — compile-verified
//
#include <hip/hip_runtime.h>
#include <math.h>

typedef float v2f __attribute__((ext_vector_type(2)));
typedef float v8f __attribute__((ext_vector_type(8)));

#define T_DIM 4096
#define BATCH 8
#define ROW_SLICES 32
#define ROWS_PER_SLICE (T_DIM / ROW_SLICES)   // 128 rows per wave
#define COL_TILES (T_DIM / 16)                // 256 tiles of 16 columns

// ---------------------------------------------------------------------------
// Kernel A: partials[slice][j] = sum_{i in slice} mask[i*T + j]
// Implemented as C += ones(16x4) * B(4x16 tile of mask) with
// V_WMMA_F32_16X16X4_F32. Every row of C equals the column partial sums.
// ---------------------------------------------------------------------------
__global__ __launch_bounds__(256) void colsum_wmma(const float* __restrict__ mask,
                                                   float* __restrict__ partials) {
  const int lane    = threadIdx.x & 31;
  const int wave_id = blockIdx.x * 8 + (threadIdx.x >> 5);  // 1024 blocks * 8 waves = 8192
  const int colTile = wave_id & (COL_TILES - 1);
  const int slice   = wave_id >> 8;                         // wave_id / COL_TILES

  // B-matrix (4x16, f32) VGPR layout: lanes 0-15 hold K=0 (b.x) and K=1 (b.y),
  // lanes 16-31 hold K=2 (b.x) and K=3 (b.y); N = lane & 15.
  const int col  = (colTile << 4) + (lane & 15);
  const int rsel = (lane >> 4) << 1;                        // 0 for lanes 0-15, 2 for 16-31

  v2f a; a.x = 1.0f; a.y = 1.0f;                            // A = all ones (16x4)
  v8f c = {};

  const float* base = mask + col;
  int row = slice * ROWS_PER_SLICE + rsel;

  #pragma unroll 4
  for (int it = 0; it < ROWS_PER_SLICE / 4; ++it) {
    v2f b;
    b.x = base[row * T_DIM];           // M[row    , col]
    b.y = base[(row + 1) * T_DIM];     // M[row + 1, col]
    // D = A*B + C  (8 args: neg_a, A, neg_b, B, c_mod, C, reuse_a, reuse_b)
    c = __builtin_amdgcn_wmma_f32_16x16x4_f32(false, a, false, b,
                                              (short)0, c, false, false);
    row += 4;
  }

  // C row M=0 lives in VGPR0 of lanes 0-15 (N = lane); all rows are identical.
  if (lane < 16) partials[slice * T_DIM + col] = c[0];
}

// ---------------------------------------------------------------------------
// Kernel B: S[j] = sum_r partials[r][j];  out[b,j] = (x[b,j]*(1+S[j]) - bias[j])*weight[j]
// ---------------------------------------------------------------------------
__global__ __launch_bounds__(256) void finalize_kernel(const float* __restrict__ x,
                                                       const float* __restrict__ weight,
                                                       const float* __restrict__ bias,
                                                       const float* __restrict__ partials,
                                                       float* __restrict__ out) {
  const int j = blockIdx.x * 256 + threadIdx.x;  // 16 blocks cover j = 0..4095
  float s = 0.0f;
  #pragma unroll
  for (int r = 0; r < ROW_SLICES; ++r) s += partials[r * T_DIM + j];

  const float scale = 1.0f + s;
  const float w  = weight[j];
  const float bv = bias[j];
  #pragma unroll
  for (int b = 0; b < BATCH; ++b) {
    const int idx = b * T_DIM + j;
    out[idx] = (x[idx] * scale - bv) * w;
  }
}

// ---------------------------------------------------------------------------
// Kernel C: ld = log_det[0] + B*(log(T-1)-log(T)) + B*sum_j log|weight[j]|
// ---------------------------------------------------------------------------
__global__ __launch_bounds__(256) void logdet_kernel(const float* __restrict__ log_det,
                                                     const float* __restrict__ weight,
                                                     float* __restrict__ out_ld) {
  __shared__ float red[256];
  float s = 0.0f;
  for (int j = threadIdx.x; j < T_DIM; j += 256) s += logf(fabsf(weight[j]));
  red[threadIdx.x] = s;
  __syncthreads();
  for (int off = 128; off > 0; off >>= 1) {
    if (threadIdx.x < off) red[threadIdx.x] += red[threadIdx.x + off];
    __syncthreads();
  }
  if (threadIdx.x == 0) {
    const float t = (float)T_DIM;
    out_ld[0] = log_det[0] + (float)BATCH * (logf(t - 1.0f) - logf(t))
                           + (float)BATCH * red[0];
  }
}

// ---------------------------------------------------------------------------
extern "C" void kernel_launch(void* const* d_in, const int* in_sizes, int n_in,
                              void* d_out, int out_size, void* d_ws, size_t ws_size,
                              hipStream_t stream) {
  // setup_inputs() order: inputs, log_det, weight, bias, mask
  const float* x       = (const float*)d_in[0];  // (8,1,64,64)
  const float* log_det = (const float*)d_in[1];  // (1,)
  const float* weight  = (const float*)d_in[2];  // (1,64,64)
  const float* bias    = (const float*)d_in[3];  // (1,64,64)
  const float* mask    = (const float*)d_in[4];  // (4096,64,64)

  float* out      = (float*)d_out;               // 32768 outputs + 1 log-det
  float* partials = (float*)d_ws;                // 32*4096 f32 = 512 KB scratch

  colsum_wmma<<<1024, 256, 0, stream>>>(mask, partials);
  finalize_kernel<<<16, 256, 0, stream>>>(x, weight, bias, partials, out);
  logdet_kernel<<<1, 256, 0, stream>>>(log_det, weight, out + BATCH * T_DIM);
}